// Curvature_59244778881459
// MI455X (gfx1250) — compile-verified
//
#include <hip/hip_runtime.h>
#include <cstdint>

// Problem constants (match reference)
#define HH   64
#define WWID 2048
#define BB   64
#define TILE 256      // output columns per block
#define LOADN 272     // TILE + 16 slack (266 needed: 5 left halo + 256 + 5 right halo)
#define BPB  16       // batches per block (amortizes per-row constants, keeps grid at 2048 blocks)

// Fast hardware exp2 (v_exp_f32)
#if __has_builtin(__builtin_amdgcn_exp2f)
#define EXP2F(v) __builtin_amdgcn_exp2f(v)
#else
#define EXP2F(v) exp2f(v)
#endif

// Fast hardware sin/cos (v_sin_f32 / v_cos_f32 compute sin(x*2pi) on pre-scaled input).
// pitch is in [-0.43, +0.054] rad -> well inside the accurate range; avoids the
// scalarized libm Payne-Hanek blob (~450 SALU instructions) seen in round 2.
#define INV2PI 0.15915494309189535f
#if __has_builtin(__builtin_amdgcn_sinf)
#define SINF_R(v) __builtin_amdgcn_sinf((v) * INV2PI)
#else
#define SINF_R(v) sinf(v)
#endif
#if __has_builtin(__builtin_amdgcn_cosf)
#define COSF_R(v) __builtin_amdgcn_cosf((v) * INV2PI)
#else
#define COSF_R(v) cosf(v)
#endif

// --- CDNA5 async global->LDS copy (tracked by ASYNCcnt) -------------------
__device__ __forceinline__ void async_load_b32(void* lds, const float* g) {
    unsigned int       laddr = (unsigned int)(uintptr_t)lds;          // low 32 bits of generic LDS ptr = LDS byte addr
    unsigned long long gaddr = (unsigned long long)(uintptr_t)g;
    asm volatile("global_load_async_to_lds_b32 %0, %1, off"
                 :
                 : "v"(laddr), "v"(gaddr)
                 : "memory");
}
__device__ __forceinline__ void wait_asynccnt2() { asm volatile("s_wait_asynccnt 2" ::: "memory"); }
__device__ __forceinline__ void wait_asynccnt0() { asm volatile("s_wait_asynccnt 0" ::: "memory"); }

// Compile-time tap constants: cos/sin(j * 2*pi/2048), j=1..5 (FFT twiddles)
#define TC1 0.99999529380957619f
#define TC2 0.99998117528260111f
#define TC3 0.99995764455196390f
#define TC4 0.99992470183914450f
#define TC5 0.99988234745421256f
#define TS1 0.0030679567629659761f
#define TS2 0.0061358846491544753f
#define TS3 0.0092037547820598194f
#define TS4 0.0122715382857199261f
#define TS5 0.0153392062849881010f
// sin(pi/2048): 1 - cos(delta) = 2*sin^2(delta/2), computed cancellation-free
#define SHALF 0.0015339801862847655f

__global__ __launch_bounds__(TILE)
void curvature_kernel(const float* __restrict__ x, float* __restrict__ out) {
    __shared__ float ldsX[2][LOADN];   // raw input, double-buffered for async pipeline
    __shared__ float ldsD[LOADN];      // depth values (single buffer, rebuilt per batch)

    const int t  = threadIdx.x;
    const int c0 = blockIdx.x * TILE;  // first output column of this tile
    const int h  = blockIdx.y;
    const int b0 = blockIdx.z * BPB;

    // ---- per-row constants --------------------------------------------------
    const float FOVR  = 0.48869219055841229f;   // 28 deg in rad
    const float FDABS = 0.43633231299858238f;   // 25 deg in rad
    const float pitch = (1.0f - (float)h * (1.0f / 64.0f)) * FOVR - FDABS;
    const float cp = COSF_R(pitch), sp = SINF_R(pitch);
    const float cp2 = cp * cp, sp2 = sp * sp;
    const float omg     = (2.0f * SHALF * SHALF) * cp2;   // 1 - g  (g = cp^2 cosD + sp^2)
    const float two_omg = omg + omg;

    // ---- async-load indices (uniform: every wave issues exactly 2 instrs) ---
    const int i1  = t;                 // covers LDS slots 0..255
    const int i2  = 256 + (t & 15);    // covers LDS slots 256..271 (redundant lanes, same data)
    int gc1 = c0 - 5 + i1; gc1 = gc1 < 0 ? 0 : (gc1 > WWID - 1 ? WWID - 1 : gc1);
    int gc2 = c0 - 5 + i2; gc2 = gc2 < 0 ? 0 : (gc2 > WWID - 1 ? WWID - 1 : gc2);

    // issue batch b's tile into buffer `buf`
    auto issue = [&](int buf, int b) {
        const float* xrow = x + ((size_t)b * HH + h) * (size_t)WWID;
        async_load_b32(&ldsX[buf][i1], xrow + gc1);
        async_load_b32(&ldsX[buf][i2], xrow + gc2);
    };

    issue(0, b0);   // prime the pipeline

    const int w  = c0 + t;       // this thread's output column
    const int ic = t + 5;        // its center index in the haloed LDS arrays
    const bool in_inner = (w >= 5) && (w <= WWID - 7);

    for (int k = 0; k < BPB; ++k) {
        const int  b    = b0 + k;
        const int  buf  = k & 1;
        const bool more = (k + 1 < BPB);

        if (more) issue(buf ^ 1, b + 1);   // prefetch next batch into the other buffer
        if (more) wait_asynccnt2();        // oldest 2 (this batch's) have landed, in order
        else      wait_asynccnt0();
        __syncthreads();                   // all waves' async data visible; ldsD safe to rewrite

        // ---- depth phase: d = clamp(exp2(1.5x + 4.5) - 1, 1, 63); 0 outside [0,W) ----
        {
            int   gc = c0 - 5 + i1;
            float d  = 0.0f;
            if (gc >= 0 && gc < WWID) {
                float e = EXP2F(fmaf(1.5f, ldsX[buf][i1], 4.5f)) - 1.0f;
                d = fminf(fmaxf(e, 1.0f), 63.0f);
            }
            ldsD[i1] = d;
        }
        if (t < 16) {
            int   i  = 256 + t;
            int   gc = c0 - 5 + i;
            float d  = 0.0f;
            if (gc >= 0 && gc < WWID) {
                float e = EXP2F(fmaf(1.5f, ldsX[buf][i], 4.5f)) - 1.0f;
                d = fminf(fmaxf(e, 1.0f), 63.0f);
            }
            ldsD[i] = d;
        }
        __syncthreads();

        // ---- stencil phase ------------------------------------------------------
        const float dm5 = ldsD[ic - 5], dm4 = ldsD[ic - 4], dm3 = ldsD[ic - 3];
        const float dm2 = ldsD[ic - 2], dm1 = ldsD[ic - 1];
        const float d0  = ldsD[ic];
        const float dp1 = ldsD[ic + 1], dp2 = ldsD[ic + 2], dp3 = ldsD[ic + 3];
        const float dp4 = ldsD[ic + 4], dp5 = ldsD[ic + 5];

        // curvature = cp^2 (A^2 + B^2) + sp^2 C^2   (exact reduction of the 3-ch conv)
        const float p1 = dm1 + dp1, p2 = dm2 + dp2, p3 = dm3 + dp3;
        const float p4 = dm4 + dp4, p5 = dm5 + dp5;
        const float A = fmaf(TC1, p1, fmaf(TC2, p2, fmaf(TC3, p3,
                        fmaf(TC4, p4, fmaf(TC5, p5, -10.0f * d0)))));
        const float Bv = fmaf(TS1, dp1 - dm1, fmaf(TS2, dp2 - dm2, fmaf(TS3, dp3 - dm3,
                         fmaf(TS4, dp4 - dm4, TS5 * (dp5 - dm5)))));
        const float Cv = (((p1 + p2) + (p3 + p4)) + p5) - 10.0f * d0;
        const float curv = fmaf(cp2, fmaf(A, A, Bv * Bv), sp2 * (Cv * Cv));

        // masks: diff^2 = (d0-d1)^2 + 2 d0 d1 (1-g);  diff_ratio^2 = 2(1-g) d0^2
        const float d1 = dp1;
        const float dd = d0 - d1;
        float diff2 = fmaf(two_omg * d0, d1, dd * dd);
        if (w == WWID - 1) diff2 = 0.0f;               // reference pads diff with 0 at end
        const bool mask1 = diff2 > 0.01f;
        const bool mask2 = two_omg * (d0 * d0) < 0.01f;
        const float dis  = d0 * d0;
        const float thr  = 2e-4f * dis;
        const bool outlier = diff2 > thr * thr;        // diff > 2e-4*dis, both sides >= 0
        const bool picked  = (in_inner && mask1 && mask2) || outlier;

        out[((size_t)b * HH + h) * (size_t)WWID + w] = picked ? __builtin_inff() : curv;
        // next iteration's __syncthreads (after the wait) guards ldsD reuse
    }
}

extern "C" void kernel_launch(void* const* d_in, const int* in_sizes, int n_in,
                              void* d_out, int out_size, void* d_ws, size_t ws_size,
                              hipStream_t stream) {
    const float* x  = (const float*)d_in[0];
    float*      out = (float*)d_out;
    dim3 grid(WWID / TILE, HH, BB / BPB);   // 8 x 64 x 4 = 2048 blocks, 256 threads (8 waves) each
    curvature_kernel<<<grid, TILE, 0, stream>>>(x, out);
}